// HashEmbedder_28200755266114
// MI455X (gfx1250) — compile-verified
//
#include <hip/hip_runtime.h>
#include <cmath>
#include <cstdint>

#define NLVL   16
#define LOG2T  19
#define TSIZE  (1u << LOG2T)
#define TMASK  (TSIZE - 1u)
#define PRIME1 2654435761u
#define PRIME2 805459861u
#define BLOCK  256

typedef float vfloat4 __attribute__((ext_vector_type(4)));

struct Res16 { float r[NLVL]; };

__global__ __launch_bounds__(BLOCK) void hash_embed_kernel(
    const float* __restrict__ x,
    const float* __restrict__ tables,
    float* __restrict__ out,
    Res16 lp)
{
    __shared__ float sx[BLOCK * 3];        // 3 KB staged coords (async-filled)
    __shared__ float sout[32 * 257];       // ~32.1 KB, pad 257 -> bank-conflict free

    const int tid = threadIdx.x;
    const long long pbase = (long long)blockIdx.x * BLOCK;

    // ---- CDNA5 async global->LDS: stage 256 float3 coords (3072 B) ----
    // 192 lanes each move one b128 chunk; tracked on ASYNCcnt, no VGPR data.
    if (tid < (BLOCK * 12) / 16) {
        unsigned lds = (unsigned)(uintptr_t)(&sx[0]) + (unsigned)tid * 16u;
        unsigned long long ga =
            (unsigned long long)(uintptr_t)(x + pbase * 3) + (unsigned long long)tid * 16u;
        asm volatile("global_load_async_to_lds_b128 %0, %1, off"
                     :: "v"(lds), "v"(ga)
                     : "memory");
    }
    asm volatile("s_wait_asynccnt 0" ::: "memory");
    __syncthreads();

    float cx = sx[tid * 3 + 0];
    float cy = sx[tid * 3 + 1];
    float cz = sx[tid * 3 + 2];
    cx = fminf(fmaxf(cx, -1.0f), 1.0f);
    cy = fminf(fmaxf(cy, -1.0f), 1.0f);
    cz = fminf(fmaxf(cz, -1.0f), 1.0f);

    #pragma unroll
    for (int l = 0; l < NLVL; ++l) {
        const float res  = lp.r[l];
        const float hres = 0.5f * res;     // 1/grid_size
        const float gs   = 2.0f / res;     // grid_size

        float tx = (cx + 1.0f) * hres;
        float ty = (cy + 1.0f) * hres;
        float tz = (cz + 1.0f) * hres;
        float fx = floorf(tx), fy = floorf(ty), fz = floorf(tz);
        int   bx = (int)fx,    by = (int)fy,    bz = (int)fz;
        float wx = fminf(fmaxf((cx - (fx * gs - 1.0f)) * hres, 0.0f), 1.0f);
        float wy = fminf(fmaxf((cy - (fy * gs - 1.0f)) * hres, 0.0f), 1.0f);
        float wz = fminf(fmaxf((cz - (fz * gs - 1.0f)) * hres, 0.0f), 1.0f);

        // int32-wraparound hash, shared partial products across the 8 corners
        unsigned hx0 = (unsigned)bx,          hx1 = hx0 + 1u;
        unsigned hy0 = (unsigned)by * PRIME1, hy1 = hy0 + PRIME1;
        unsigned hz0 = (unsigned)bz * PRIME2, hz1 = hz0 + PRIME2;

        const float2* __restrict__ tb = (const float2*)tables + (size_t)l * TSIZE;
        // issue all 8 gathers up front (MLP); L2-resident table
        float2 e000 = tb[(hx0 ^ hy0 ^ hz0) & TMASK];
        float2 e001 = tb[(hx0 ^ hy0 ^ hz1) & TMASK];
        float2 e010 = tb[(hx0 ^ hy1 ^ hz0) & TMASK];
        float2 e011 = tb[(hx0 ^ hy1 ^ hz1) & TMASK];
        float2 e100 = tb[(hx1 ^ hy0 ^ hz0) & TMASK];
        float2 e101 = tb[(hx1 ^ hy0 ^ hz1) & TMASK];
        float2 e110 = tb[(hx1 ^ hy1 ^ hz0) & TMASK];
        float2 e111 = tb[(hx1 ^ hy1 ^ hz1) & TMASK];

        float ux = 1.0f - wx, uy = 1.0f - wy, uz = 1.0f - wz;
        float w00 = ux * uy, w01 = ux * wy, w10 = wx * uy, w11 = wx * wy;
        float w000 = w00 * uz, w001 = w00 * wz;
        float w010 = w01 * uz, w011 = w01 * wz;
        float w100 = w10 * uz, w101 = w10 * wz;
        float w110 = w11 * uz, w111 = w11 * wz;

        float f0 = w000 * e000.x + w001 * e001.x + w010 * e010.x + w011 * e011.x
                 + w100 * e100.x + w101 * e101.x + w110 * e110.x + w111 * e111.x;
        float f1 = w000 * e000.y + w001 * e001.y + w010 * e010.y + w011 * e011.y
                 + w100 * e100.y + w101 * e101.y + w110 * e110.y + w111 * e111.y;

        // feature-major staging: lanes write consecutive addresses -> no conflicts
        sout[(2 * l + 0) * 257 + tid] = f0;
        sout[(2 * l + 1) * 257 + tid] = f1;
    }
    __syncthreads();

    // ---- coalesced non-temporal writeback: 8 x b128 per thread ----
    // NT keeps the 128 MB output stream from evicting the 64 MB of tables in L2.
    vfloat4* __restrict__ ob = (vfloat4*)(out + pbase * 32);
    #pragma unroll
    for (int i = 0; i < 8; ++i) {
        int q  = tid + i * BLOCK;     // float4 index within block's 32 KB slab
        int pl = q >> 3;              // local point
        int fb = (q & 7) << 2;        // feature base
        vfloat4 v;
        v.x = sout[(fb + 0) * 257 + pl];
        v.y = sout[(fb + 1) * 257 + pl];
        v.z = sout[(fb + 2) * 257 + pl];
        v.w = sout[(fb + 3) * 257 + pl];
        __builtin_nontemporal_store(v, ob + q);
    }
}

extern "C" void kernel_launch(void* const* d_in, const int* in_sizes, int n_in,
                              void* d_out, int out_size, void* d_ws, size_t ws_size,
                              hipStream_t stream) {
    const float* x      = (const float*)d_in[0];
    const float* tables = (const float*)d_in[1];
    float* out          = (float*)d_out;

    const int npts = in_sizes[0] / 3;           // 1,048,576 (divisible by 256)
    Res16 lp;
    const double bg = exp((log(512.0) - log(16.0)) / 15.0);
    for (int i = 0; i < NLVL; ++i)
        lp.r[i] = (float)floor(16.0 * pow(bg, (double)i));

    const int blocks = npts / BLOCK;            // exact: 4096 full blocks
    hash_embed_kernel<<<blocks, BLOCK, 0, stream>>>(x, tables, out, lp);
}